// MLP_6502580486167
// MI455X (gfx1250) — compile-verified
//
#include <hip/hip_runtime.h>
#include <hip/hip_bf16.h>

typedef __attribute__((ext_vector_type(2))) float v2f;
typedef __attribute__((ext_vector_type(8))) float v8f;

#define C_IN  128
#define C_HID 64
#define C_OUT 32

// ---------------------------------------------------------------- utilities
__global__ void k_fill(float* __restrict__ p, float v, int n) {
  int i = blockIdx.x * blockDim.x + threadIdx.x;
  if (i < n) p[i] = v;
}

__global__ void k_deg_accum(const int* __restrict__ dst, float* __restrict__ deg, int E) {
  int e = blockIdx.x * blockDim.x + threadIdx.x;
  if (e < E) atomicAdd(&deg[dst[e]], 1.0f);
}

__global__ void k_rsqrt_inplace(float* __restrict__ deg, int n) {
  int i = blockIdx.x * blockDim.x + threadIdx.x;
  if (i < n) deg[i] = rsqrtf(deg[i]);   // deg >= 1 due to self-loops
}

// ------------------------------------------------- fp32 WMMA GEMM: C = A@B
// One wave computes one 16x16 tile of C. K stepped by 4 via
// V_WMMA_F32_16X16X4_F32 (exact fp32 matrix pipe; GEMM is not the bottleneck,
// so no precision-losing downcast).
// A-fragment (16x4, 32-bit): lanes 0-15: M=lane, v0=K+0, v1=K+1;
//                            lanes 16-31: M=lane-16, v0=K+2, v1=K+3.
// B-fragment (4x16): lanes 0-15: N=lane, v0=K+0 row, v1=K+1 row;
//                    lanes 16-31: N=lane-16, v0=K+2, v1=K+3.
// D (16x16 f32, 8 VGPRs): VGPR r -> M = r + 8*(lane>=16), N = lane%16.
__global__ void k_gemm_wmma_f32(const float* __restrict__ A,
                                const float* __restrict__ B,
                                float* __restrict__ C,
                                int M, int K, int N) {
  int wave = (blockIdx.x * blockDim.x + threadIdx.x) >> 5;
  int lane = threadIdx.x & 31;
  int ntiles = N >> 4;
  int mt = wave / ntiles;
  int nt = wave - mt * ntiles;
  if (mt * 16 >= M) return;            // uniform per wave: EXEC stays all-1s

  int lrow  = lane & 15;
  int khalf = (lane >> 4) << 1;        // 0 or 2

  const float* arow = A + (size_t)(mt * 16 + lrow) * K;
  const float* bcol = B + (nt * 16 + lrow);

  v8f acc = {};
  for (int k = 0; k < K; k += 4) {
    v2f a, b;
    a.x = arow[k + khalf];
    a.y = arow[k + khalf + 1];
    b.x = bcol[(size_t)(k + khalf) * N];
    b.y = bcol[(size_t)(k + khalf + 1) * N];
    acc = __builtin_amdgcn_wmma_f32_16x16x4_f32(
        /*neg_a=*/false, a, /*neg_b=*/false, b,
        /*c_mod=*/(short)0, acc, /*reuse_a=*/false, /*reuse_b=*/false);
  }

  int rbase = (lane >> 4) << 3;        // 0 or 8
  float* crow = C + (size_t)(mt * 16 + rbase) * N + nt * 16 + lrow;
#pragma unroll
  for (int r = 0; r < 8; ++r) crow[(size_t)r * N] = acc[r];
}

// --------------------------------------------- edge scatter: agg[d] += w*h[s]
// tpe = C/4 threads per edge, float4 gather + 4 f32 atomics (L2-resident).
__global__ void k_scatter(const int* __restrict__ src, const int* __restrict__ dst,
                          const float* __restrict__ dinv, const float* __restrict__ h,
                          float* __restrict__ agg, int E, int cshift) {
  int gid = blockIdx.x * blockDim.x + threadIdx.x;
  int eshift = cshift - 2;                     // log2(threads per edge)
  int e = gid >> eshift;
  if (e >= E) return;
  int c = (gid & ((1 << eshift) - 1)) << 2;
  int s = src[e];
  int d = dst[e];
  float w = dinv[s] * dinv[d];
  const float4 v = *(const float4*)(h + ((size_t)s << cshift) + c);
  float* ap = agg + ((size_t)d << cshift) + c;
  atomicAdd(ap + 0, w * v.x);
  atomicAdd(ap + 1, w * v.y);
  atomicAdd(ap + 2, w * v.z);
  atomicAdd(ap + 3, w * v.w);
}

// ------------------- add self-loop term + bias (+ optional ReLU), in place
__global__ void k_finalize(float* __restrict__ seg, const float* __restrict__ hlin,
                           const float* __restrict__ dinv, const float* __restrict__ bias,
                           int total, int cshift, int do_relu) {
  int idx = blockIdx.x * blockDim.x + threadIdx.x;
  if (idx >= total) return;
  int i = idx >> cshift;
  int c = idx & ((1 << cshift) - 1);
  float di = dinv[i];
  float v = seg[idx] + di * di * hlin[idx] + bias[c];
  if (do_relu) v = fmaxf(v, 0.0f);
  seg[idx] = v;
}

// ---------------------------------------------------------------- launcher
extern "C" void kernel_launch(void* const* d_in, const int* in_sizes, int n_in,
                              void* d_out, int out_size, void* d_ws, size_t ws_size,
                              hipStream_t stream) {
  const float* x  = (const float*)d_in[0];
  const int*   ei = (const int*)d_in[1];
  const float* W1 = (const float*)d_in[2];
  const float* b1 = (const float*)d_in[3];
  const float* W2 = (const float*)d_in[4];
  const float* b2 = (const float*)d_in[5];

  const int N = in_sizes[0] / C_IN;       // 100000 (multiple of 16)
  const int E = in_sizes[1] / 2;          // 1600000
  const int* src = ei;                    // edge_index[0]
  const int* dst = ei + E;                // edge_index[1]

  // d_out = [out (N*32)] ++ [h (N*64)]
  float* outseg = (float*)d_out;
  float* hseg   = outseg + (size_t)N * C_OUT;

  // workspace: dinv | h1 | h2   (~38.8 MB)
  float* ws   = (float*)d_ws;
  float* dinv = ws;
  float* h1   = dinv + N;
  float* h2   = h1 + (size_t)N * C_HID;

  const int B = 256;
  const int total_out = N * (C_OUT + C_HID);

  // zero output accumulators; deg starts at 1 (self-loop)
  k_fill<<<(total_out + B - 1) / B, B, 0, stream>>>((float*)d_out, 0.0f, total_out);
  k_fill<<<(N + B - 1) / B, B, 0, stream>>>(dinv, 1.0f, N);
  k_deg_accum<<<(E + B - 1) / B, B, 0, stream>>>(dst, dinv, E);
  k_rsqrt_inplace<<<(N + B - 1) / B, B, 0, stream>>>(dinv, N);

  // layer 1: h1 = x @ W1  (6250 x 4 tiles -> 25000 waves)
  {
    int threads = (N / 16) * (C_HID / 16) * 32;
    k_gemm_wmma_f32<<<(threads + B - 1) / B, B, 0, stream>>>(x, W1, h1, N, C_IN, C_HID);
  }
  {
    int threads = E * (C_HID / 4);
    k_scatter<<<(threads + B - 1) / B, B, 0, stream>>>(src, dst, dinv, h1, hseg, E, 6);
  }
  k_finalize<<<(N * C_HID + B - 1) / B, B, 0, stream>>>(hseg, h1, dinv, b1, N * C_HID, 6, 1);

  // layer 2: h2 = h @ W2  (6250 x 2 tiles -> 12500 waves)
  {
    int threads = (N / 16) * (C_OUT / 16) * 32;
    k_gemm_wmma_f32<<<(threads + B - 1) / B, B, 0, stream>>>(hseg, W2, h2, N, C_HID, C_OUT);
  }
  {
    int threads = E * (C_OUT / 4);
    k_scatter<<<(threads + B - 1) / B, B, 0, stream>>>(src, dst, dinv, h2, outseg, E, 5);
  }
  k_finalize<<<(N * C_OUT + B - 1) / B, B, 0, stream>>>(outseg, h2, dinv, b2, N * C_OUT, 5, 0);
}